// TurboQuantKVCache_66125316489462
// MI455X (gfx1250) — compile-verified
//
#include <hip/hip_runtime.h>
#include <stdint.h>

// ---------------------------------------------------------------------------
// TurboQuantKVCache fused quantize->dequantize for MI455X (gfx1250, wave32).
//   B=4 H=8 S=4096 D=128.  Streaming, HBM-bound (~256 MiB -> ~11.5 us @23.3TB/s).
//   One wave32 per row of 128 f32; lane i owns elements [4i..4i+3].
//   Input staged through LDS with GLOBAL_LOAD_ASYNC_TO_LDS_B128 (ASYNCcnt),
//   double-buffered per wave; NT temporal hints on the streaming traffic
//   (read-once inputs, write-once outputs; working set > 192MB L2).
// ---------------------------------------------------------------------------

typedef float v4f __attribute__((ext_vector_type(4)));

namespace {

constexpr int kD            = 128;
constexpr int kS            = 4096;
constexpr int kBH           = 4 * 8;                 // B*H
constexpr int kRowsPerTen   = kBH * kS;              // 131072 rows per tensor
constexpr int kRows         = 2 * kRowsPerTen;       // k rows then v rows
constexpr float kSqrtD      = 11.313708498984761f;   // sqrt(128)
constexpr int kWavesPerBlk  = 8;                     // 256 threads
constexpr int kBlocks       = 2048;                  // 16384 persistent waves

} // namespace

// Wave32 butterfly sum: all lanes end with the full 128-element total.
__device__ __forceinline__ float wave_sum(float v) {
#pragma unroll
  for (int m = 1; m < 32; m <<= 1) v += __shfl_xor(v, m, 32);
  return v;
}

// Fused searchsorted(side='left') + centroid lookup, all constants folded.
// idx = #{ boundaries[j] < x }  ==>  q = (x > b_j) ? c[j+1] : q  (15 steps,
// v_cmp_gt_f32 + v_cndmask_b32, dual-issue friendly; equality at a boundary
// falls to the lower bucket, matching side='left').
__device__ __forceinline__ float quant_dequant(float xn) {
  constexpr float C[16] = {-2.7326f, -2.069f, -1.618f, -1.2562f,
                           -0.9423f, -0.6568f, -0.3881f, -0.1284f,
                            0.1284f,  0.3881f,  0.6568f,  0.9423f,
                            1.2562f,  1.618f,   2.069f,   2.7326f};
  float q = C[0];
#pragma unroll
  for (int j = 0; j < 15; ++j) {
    const float b = (C[j] + C[j + 1]) * 0.5f;   // f32, matches jnp boundary calc
    q = (xn > b) ? C[j + 1] : q;
  }
  return q;
}

// Async copy of this wave's 512B row into an LDS staging buffer.
// GV mode: vdst = LDS byte address (low 32 bits of generic shared ptr),
// vaddr = 64-bit global address, no SADDR. Tracked by ASYNCcnt.
// NT hint: data is read exactly once -> don't retain in cache hierarchy.
__device__ __forceinline__ void async_stage_row(const float* __restrict__ src,
                                                v4f* lds_buf, int lane) {
  const v4f* g = reinterpret_cast<const v4f*>(src) + lane;
  uint32_t l = (uint32_t)(uintptr_t)(lds_buf + lane);
  asm volatile("global_load_async_to_lds_b128 %0, %1, off th:TH_LOAD_NT"
               :: "v"(l), "v"(g) : "memory");
}

__global__ void __launch_bounds__(256)
turboquant_kv_qdq(const int* __restrict__ pos,
                  const float* __restrict__ kval,
                  const float* __restrict__ vval,
                  float* __restrict__ out) {
  __shared__ v4f stage[2][kWavesPerBlk][32];   // 8 KB: 2 bufs x 512B x 8 waves

  const int lane  = threadIdx.x & 31;
  const int wave  = threadIdx.x >> 5;
  const int nglob = (gridDim.x * blockDim.x) >> 5;          // total waves
  int row = (blockIdx.x * blockDim.x + threadIdx.x) >> 5;   // wave-uniform

  if (row >= kRows) return;

  auto src_ptr = [&](int r) -> const float* {
    return (r < kRowsPerTen) ? (kval + (size_t)r * kD)
                             : (vval + (size_t)(r - kRowsPerTen) * kD);
  };

  // Prologue: stage first row into buffer 0.
  async_stage_row(src_ptr(row), &stage[0][wave][0], lane);

  int buf = 0;
  for (;;) {
    const int next = row + nglob;
    if (next < kRows) {
      // Prefetch next row into the other buffer, then wait for current row.
      async_stage_row(src_ptr(next), &stage[buf ^ 1][wave][0], lane);
      asm volatile("s_wait_asynccnt 1" ::: "memory");  // in-order: current done
    } else {
      asm volatile("s_wait_asynccnt 0" ::: "memory");
    }

    const v4f x = stage[buf][wave][lane];               // ds_load_b128

    // mean over 128 elements
    const float mean = wave_sum(x[0] + x[1] + x[2] + x[3]) * (1.0f / 128.0f);
    const float cx = x[0] - mean, cy = x[1] - mean;
    const float cz = x[2] - mean, cw = x[3] - mean;
    // centered L2 norm, clamped like the reference
    const float ssq = wave_sum(cx * cx + cy * cy + cz * cz + cw * cw);
    const float mag = fmaxf(sqrtf(ssq), 1e-8f);

    const float s  = kSqrtD / mag;        // x_norm = xc * s
    const float os = mag * (1.0f / kSqrtD);

    v4f o;
    o[0] = fmaf(quant_dequant(cx * s), os, mean);
    o[1] = fmaf(quant_dequant(cy * s), os, mean);
    o[2] = fmaf(quant_dequant(cz * s), os, mean);
    o[3] = fmaf(quant_dequant(cw * s), os, mean);

    // Scatter row to out[..., input_pos[s_in], :] (identity permutation here,
    // so every output row is rewritten each call -> safe under replay).
    const bool is_v = (row >= kRowsPerTen);
    const int  rr   = is_v ? row - kRowsPerTen : row;
    const int  bh   = rr >> 12;            // / kS
    const int  sin  = rr & (kS - 1);
    const int  sout = pos[sin];            // wave-uniform -> scalar load
    float* dst = out + (is_v ? (size_t)kRowsPerTen * kD : 0)
                     + ((size_t)bh * kS + (size_t)sout) * kD;
    // Write-once stream: single b128 store with NT hint (write-rinse, no
    // L2 retention of data nothing will re-read).
    __builtin_nontemporal_store(o, reinterpret_cast<v4f*>(dst) + lane);

    if (next >= kRows) break;
    row = next;
    buf ^= 1;
  }
}

extern "C" void kernel_launch(void* const* d_in, const int* in_sizes, int n_in,
                              void* d_out, int out_size, void* d_ws, size_t ws_size,
                              hipStream_t stream) {
  (void)in_sizes; (void)n_in; (void)out_size; (void)d_ws; (void)ws_size;
  const int*   pos = (const int*)d_in[0];
  const float* k   = (const float*)d_in[1];
  const float* v   = (const float*)d_in[2];
  float*       out = (float*)d_out;
  turboquant_kv_qdq<<<dim3(kBlocks), dim3(256), 0, stream>>>(pos, k, v, out);
}